// AlignedAttention_48034914238598
// MI455X (gfx1250) — compile-verified
//
#include <hip/hip_runtime.h>
#include <hip/hip_bf16.h>

// AlignedAttention (align=False): 2x2-patch gather.
//   out[b,c,2py+ky,2px+kx] = value[b,c,2sy+ky,2sx+kx],  s=index_map[b,py*96+px]
// Pure data movement (302 MB total) -> bandwidth bound, ~13us at 23.3 TB/s.
// CDNA5 paths used: async global->LDS index staging (ASYNCcnt),
// non-temporal output stores (preserve 192MB L2 for the gather source),
// global_prefetch of next channel's gather address.

typedef float f32x2 __attribute__((ext_vector_type(2)));   // native vector: OK for nontemporal builtins

constexpr int Bc      = 8;
constexpr int Cc      = 128;
constexpr int PW      = 96;           // patches per row/col
constexpr int P_TOTAL = PW * PW;      // 9216 patches per batch
constexpr int W2      = 192;          // output width (floats)
constexpr int PLANE2  = (192 * 192) / 2;   // plane size in f32x2 units = 18432
constexpr int TILE_P  = 128;          // patches per block
constexpr int N_TILES = P_TOTAL / TILE_P;  // 72
constexpr int BLOCK   = 256;
constexpr int C_GROUPS = 4;
constexpr int C_PER   = Cc / C_GROUPS;     // 32 channels per block

__global__ __launch_bounds__(BLOCK)
void AlignedAttention_gather_kernel(const int* __restrict__ index_map,
                                    const float* __restrict__ value,
                                    float* __restrict__ out) {
    __shared__ int s_idx[TILE_P];     // raw gather indices (async landing zone)
    __shared__ int s_src[TILE_P];     // src patch offset, f32x2 units (ky=0)
    __shared__ int s_dst[TILE_P];     // dst patch offset, f32x2 units (ky=0)

    const int tile = blockIdx.x % N_TILES;
    const int b    = blockIdx.x / N_TILES;
    const int cg   = blockIdx.y;
    const int t    = threadIdx.x;
    const int p0   = tile * TILE_P;

    // ---- Stage this tile's indices into LDS via CDNA5 async global->LDS copy.
    if (t < TILE_P) {
        unsigned ldsOff = (unsigned)(uintptr_t)(&s_idx[t]);   // low 32 bits = LDS byte addr
        const int* gsrc = index_map + (size_t)b * P_TOTAL + (p0 + t);
        asm volatile("global_load_async_to_lds_b32 %0, %1, off"
                     :
                     : "v"(ldsOff), "v"(gsrc)
                     : "memory");
    }
    asm volatile("s_wait_asynccnt 0" ::: "memory");
    __syncthreads();

    // ---- Precompute per-patch src/dst offsets once (reused by 32 channels x 2 rows).
    if (t < TILE_P) {
        const int p  = p0 + t;
        const int s  = s_idx[t];
        const int sy = s / PW, sx = s - sy * PW;
        const int py = p / PW, px = p - py * PW;
        // element offset (2*sy)*192 + 2*sx is even -> f32x2-unit offset:
        s_src[t] = sy * W2 + sx;
        s_dst[t] = py * W2 + px;
    }
    __syncthreads();

    // ---- Main copy loop: thread = (patch-in-tile, ky); iterate channels.
    const int pl = t & (TILE_P - 1);
    const int ky = t >> 7;                       // 0 or 1
    const int srcH = s_src[pl] + ky * (W2 / 2);  // +96 per ky (f32x2 units)
    const int dstH = s_dst[pl] + ky * (W2 / 2);

    const size_t baseB = (size_t)b * Cc * PLANE2;
    const f32x2* __restrict__ vsrc = (const f32x2*)value + baseB;
    f32x2* __restrict__       vdst = (f32x2*)out + baseB;

    const int c0 = cg * C_PER;
    #pragma unroll 8
    for (int i = 0; i < C_PER; ++i) {
        const int c = c0 + i;
        const f32x2* sp = vsrc + (size_t)c * PLANE2 + srcH;
        if (i + 1 < C_PER) {
            // next channel's gather address is known now -> global_prefetch
            __builtin_prefetch((const void*)(sp + PLANE2), 0, 3);
        }
        f32x2 v = *sp;                                    // 8B gathered load (L2-resident)
        // write-once output: non-temporal, keep L2 for the gather source
        __builtin_nontemporal_store(v, vdst + (size_t)c * PLANE2 + dstH);
    }
}

extern "C" void kernel_launch(void* const* d_in, const int* in_sizes, int n_in,
                              void* d_out, int out_size, void* d_ws, size_t ws_size,
                              hipStream_t stream) {
    // setup_inputs order: lr(0, unused), ref(1, unused), index_map(2), value(3), name(4, unused)
    const int*   index_map = (const int*)d_in[2];
    const float* value     = (const float*)d_in[3];
    float*       out       = (float*)d_out;

    dim3 grid(Bc * N_TILES, C_GROUPS);   // (576, 4) = 2304 blocks
    AlignedAttention_gather_kernel<<<grid, BLOCK, 0, stream>>>(index_map, value, out);
}